// SubGraph_89172111000347
// MI455X (gfx1250) — compile-verified
//
#include <hip/hip_runtime.h>
#include <cstdint>

typedef __attribute__((ext_vector_type(2))) float v2f;
typedef __attribute__((ext_vector_type(8))) float v8f;

#define HID 64

// ---------------- degree / norm (once, reused by all 3 layers) ----------------
__global__ void k_zero_deg(int* __restrict__ deg, int N) {
  int i = blockIdx.x * blockDim.x + threadIdx.x;
  if (i < N) deg[i] = 0;
}

__global__ void k_deg(const int* __restrict__ ei, int* __restrict__ deg, int E) {
  int e = blockIdx.x * blockDim.x + threadIdx.x;
  if (e < E) atomicAdd(&deg[ei[(size_t)E + e]], 1);
}

__global__ void k_dinv(const int* __restrict__ deg, float* __restrict__ dinv, int N) {
  int i = blockIdx.x * blockDim.x + threadIdx.x;
  if (i < N) dinv[i] = rsqrtf((float)(deg[i] + 1));   // +1 self loop; always > 0
}

// ---------------- hs = (X @ W) * dinv[row]; acc = hs (self-loop init) ----------------
__global__ __launch_bounds__(256) void k_feat(const float* __restrict__ X,
                                              const float* __restrict__ W,
                                              const float* __restrict__ dinv,
                                              float* __restrict__ hs,
                                              float* __restrict__ acc,
                                              int N, int din) {
  __shared__ float sW[24 * HID];
  int tot = din * HID;
  for (int i = threadIdx.x; i < tot; i += blockDim.x) sW[i] = W[i];
  __syncthreads();
  int n = blockIdx.x * 4 + (threadIdx.x >> 6);
  int f = threadIdx.x & 63;
  if (n >= N) return;
  const float* xr = X + (size_t)n * din;
  float s = 0.f;
  for (int k = 0; k < din; ++k) s += xr[k] * sW[k * HID + f];
  s *= dinv[n];
  size_t o = (size_t)n * HID + f;
  hs[o] = s;
  acc[o] = s;
}

// ---------------- edge scatter: acc[dst] += hs[src], 16 lanes x float4 per edge --------
__global__ __launch_bounds__(256) void k_scatter(const int* __restrict__ ei,
                                                 const float* __restrict__ hs,
                                                 float* __restrict__ acc, int E) {
  long long tid = (long long)blockIdx.x * blockDim.x + threadIdx.x;
  int e = (int)(tid >> 4);
  if (e >= E) return;
  int q = ((int)tid & 15) * 4;
  int s = ei[e];
  int d = ei[(size_t)E + e];
  const float4 v = *(const float4*)(hs + (size_t)s * HID + q);
  float* ap = acc + (size_t)d * HID + q;
  atomicAdd(ap + 0, v.x);
  atomicAdd(ap + 1, v.y);
  atomicAdd(ap + 2, v.z);
  atomicAdd(ap + 3, v.w);
}

// ---------------- per-node: relu(dinv*acc + b) -> LayerNorm(64) (one wave per node) ----
__global__ __launch_bounds__(256) void k_postln(const float* __restrict__ acc,
                                                const float* __restrict__ dinv,
                                                const float* __restrict__ bias,
                                                const float* __restrict__ g,
                                                const float* __restrict__ be,
                                                float* __restrict__ ln, int N) {
  int tid = blockIdx.x * blockDim.x + threadIdx.x;
  int n = tid >> 5, lane = tid & 31;
  if (n >= N) return;
  float dv = dinv[n];
  size_t o = (size_t)n * HID + lane;
  float a0 = fmaxf(dv * acc[o] + bias[lane], 0.f);
  float a1 = fmaxf(dv * acc[o + 32] + bias[lane + 32], 0.f);
  float s = a0 + a1, ss = a0 * a0 + a1 * a1;
  for (int m = 16; m >= 1; m >>= 1) {
    s  += __shfl_xor(s, m, 32);
    ss += __shfl_xor(ss, m, 32);
  }
  float mu  = s * (1.f / 64.f);
  float inv = rsqrtf(ss * (1.f / 64.f) - mu * mu + 1e-5f);
  ln[o]      = (a0 - mu) * inv * g[lane]      + be[lane];
  ln[o + 32] = (a1 - mu) * inv * g[lane + 32] + be[lane + 32];
}

// ---------------- Y = relu(X[N,64] @ L[64,DOUT] + lb) via V_WMMA_F32_16X16X4_F32 -------
// L is staged zero-padded into LDS once per block (shared by 8 waves) so the
// k-loop has no divergent guards; one wave computes a 16-row stripe across all
// ceil(DOUT/16) column tiles with the A fragment reused across tiles.
template <int DOUT>
__global__ __launch_bounds__(256) void k_linear_wmma(const float* __restrict__ X,
                                                     const float* __restrict__ Lw,
                                                     const float* __restrict__ lb,
                                                     float* __restrict__ Y, int N) {
  constexpr int NT = (DOUT + 15) / 16;
  constexpr int PC = NT * 16;                 // padded column count
  __shared__ float sL[HID * PC];              // <= 64*48*4 = 12 KB

  for (int i = threadIdx.x; i < HID * PC; i += blockDim.x) {
    int k = i / PC, c = i % PC;
    sL[i] = (c < DOUT) ? Lw[k * DOUT + c] : 0.f;
  }
  __syncthreads();                            // all waves participate before any exit

  int wave = (blockIdx.x * blockDim.x + threadIdx.x) >> 5;
  int lane = threadIdx.x & 31;
  int row0 = wave * 16;
  if (row0 >= N) return;                      // wave-uniform: EXEC all-1s for WMMA
  int half = lane >> 4;                       // 0: K+0/K+1 lanes, 1: K+2/K+3 lanes
  int l16  = lane & 15;
  int arow = row0 + l16;
  if (arow >= N) arow = N - 1;                // clamp (keeps EXEC full; stores guarded)
  const float* xr = X + (size_t)arow * HID;

  v8f c[NT];
#pragma unroll
  for (int ct = 0; ct < NT; ++ct) c[ct] = (v8f)0.f;

#pragma unroll
  for (int k0 = 0; k0 < HID; k0 += 4) {
    v2f a;                                    // A 16x4 f32: VGPR0=K(k0+2h), VGPR1=next K
    a.x = xr[k0 + 2 * half + 0];
    a.y = xr[k0 + 2 * half + 1];
#pragma unroll
    for (int ct = 0; ct < NT; ++ct) {
      int col = ct * 16 + l16;
      v2f b;                                  // B 4x16 f32: same K split by lane half
      b.x = sL[(k0 + 2 * half + 0) * PC + col];
      b.y = sL[(k0 + 2 * half + 1) * PC + col];
      c[ct] = __builtin_amdgcn_wmma_f32_16x16x4_f32(
          false, a, false, b, (short)0, c[ct], false, false);
    }
  }

#pragma unroll
  for (int ct = 0; ct < NT; ++ct) {
    int col = ct * 16 + l16;
    if (col >= DOUT) continue;
    float bb = lb[col];
#pragma unroll
    for (int r = 0; r < 8; ++r) {             // C/D: VGPR r -> rows r (lanes 0-15), r+8
      int row = row0 + r + half * 8;
      if (row < N) Y[(size_t)row * DOUT + col] = fmaxf(c[ct][r] + bb, 0.f);
    }
  }
}

// ---------------- final column max over N rows of [N,48]; all values >= 0 --------------
__global__ void k_outzero(float* __restrict__ o) {
  if (threadIdx.x < 48) o[threadIdx.x] = 0.f;
}

__global__ __launch_bounds__(256) void k_colmax(const float* __restrict__ Y,
                                                float* __restrict__ out, int N) {
  int t = threadIdx.x;
  if (t >= 240) return;
  int col = t % 48, r0 = t / 48;              // 5 rows per block-iteration
  float m = 0.f;
  for (long long n = (long long)blockIdx.x * 5 + r0; n < N; n += (long long)gridDim.x * 5)
    m = fmaxf(m, Y[n * 48 + col]);
  atomicMax((unsigned int*)(out + col), __float_as_uint(m));  // exact for floats >= 0
}

extern "C" void kernel_launch(void* const* d_in, const int* in_sizes, int n_in,
                              void* d_out, int out_size, void* d_ws, size_t ws_size,
                              hipStream_t stream) {
  const float* x  = (const float*)d_in[0];
  const int*   ei = (const int*)d_in[1];
  int N = in_sizes[0] / 6;
  int E = in_sizes[1] / 2;

  const float *W[3], *b[3], *g[3], *be[3], *Lw[3], *lb[3];
  for (int i = 0; i < 3; ++i) {
    W[i]  = (const float*)d_in[2 + 6 * i + 0];
    b[i]  = (const float*)d_in[2 + 6 * i + 1];
    g[i]  = (const float*)d_in[2 + 6 * i + 2];
    be[i] = (const float*)d_in[2 + 6 * i + 3];
    Lw[i] = (const float*)d_in[2 + 6 * i + 4];
    lb[i] = (const float*)d_in[2 + 6 * i + 5];
  }

  float* ws   = (float*)d_ws;
  float* dinv = ws;                               // N
  int*   deg  = (int*)(ws + N);                   // N
  float* hs   = ws + 2 * (size_t)N;               // N*64
  float* acc  = hs + (size_t)N * HID;             // N*64
  float* ln   = hs;                               // alias: hs dead after scatter
  float* bufA = acc + (size_t)N * HID;            // N*48
  float* bufB = bufA + (size_t)N * 48;            // N*48

  int nb  = (N + 255) / 256;
  int eb  = (E + 255) / 256;
  int fb  = (N + 3) / 4;
  long long sct = (long long)E * 16;
  unsigned sb = (unsigned)((sct + 255) / 256);
  int pb  = (N + 7) / 8;
  int waves = (N + 15) / 16;
  int wb  = (waves + 7) / 8;

  // structure-only degree normalization (shared by all layers)
  k_zero_deg<<<nb, 256, 0, stream>>>(deg, N);
  k_deg<<<eb, 256, 0, stream>>>(ei, deg, E);
  k_dinv<<<nb, 256, 0, stream>>>(deg, dinv, N);

  // ---- layer 1: x[N,6] -> bufA[N,12]
  k_feat<<<fb, 256, 0, stream>>>(x, W[0], dinv, hs, acc, N, 6);
  k_scatter<<<sb, 256, 0, stream>>>(ei, hs, acc, E);
  k_postln<<<pb, 256, 0, stream>>>(acc, dinv, b[0], g[0], be[0], ln, N);
  k_linear_wmma<12><<<wb, 256, 0, stream>>>(ln, Lw[0], lb[0], bufA, N);

  // ---- layer 2: bufA[N,12] -> bufB[N,24]
  k_feat<<<fb, 256, 0, stream>>>(bufA, W[1], dinv, hs, acc, N, 12);
  k_scatter<<<sb, 256, 0, stream>>>(ei, hs, acc, E);
  k_postln<<<pb, 256, 0, stream>>>(acc, dinv, b[1], g[1], be[1], ln, N);
  k_linear_wmma<24><<<wb, 256, 0, stream>>>(ln, Lw[1], lb[1], bufB, N);

  // ---- layer 3: bufB[N,24] -> bufA[N,48]
  k_feat<<<fb, 256, 0, stream>>>(bufB, W[2], dinv, hs, acc, N, 24);
  k_scatter<<<sb, 256, 0, stream>>>(ei, hs, acc, E);
  k_postln<<<pb, 256, 0, stream>>>(acc, dinv, b[2], g[2], be[2], ln, N);
  k_linear_wmma<48><<<wb, 256, 0, stream>>>(ln, Lw[2], lb[2], bufA, N);

  // ---- global max over nodes -> [48]
  k_outzero<<<1, 64, 0, stream>>>((float*)d_out);
  k_colmax<<<200, 256, 0, stream>>>(bufA, (float*)d_out, N);
}